// Repara_PhC_1x1_78408922955845
// MI455X (gfx1250) — compile-verified
//
#include <hip/hip_runtime.h>

// Reference: BOX=20, RES=200 -> N=4001 grid, 96 Gaussian holes,
// out = sigmoid((Z-0.5)/T), Z separable => rank-96 GEMM:
//   Z = Ex (4001x96) * Ey^T (96x4001), K=96 via V_WMMA_F32_16X16X4_F32.
#define NGRID   4001
#define NHOLES  96
#define KSTEPS  (NHOLES / 4)          // 24 K-steps of 4
#define MACRO   64                    // 64x64 macrotile per wave (4x4 WMMA tiles)
#define GRIDT   63                    // ceil(4001/64)
#define TOTAL_MACRO (GRIDT * GRIDT)   // 3969 waves
#define WAVES_PER_BLOCK 8

typedef __attribute__((ext_vector_type(2))) float v2f;
typedef __attribute__((ext_vector_type(8))) float v8f;

// ---------------------------------------------------------------------------
// Kernel 1: separable Gaussian tables.
//   Ex[i*96+h] = exp(-(x_i - x0_h)^2 / (2 sg^2)),  Ey likewise with y0_h.
// ---------------------------------------------------------------------------
__global__ __launch_bounds__(256) void build_tables_kernel(
    const float* __restrict__ holes,
    float* __restrict__ Ex,
    float* __restrict__ Ey)
{
    int idx = blockIdx.x * blockDim.x + threadIdx.x;
    if (idx >= NGRID * NHOLES) return;
    int i = idx / NHOLES;
    int h = idx - i * NHOLES;

    float p  = -10.0f + (float)i * 0.005f;
    float x0 = holes[h * 3 + 0];
    float y0 = holes[h * 3 + 1];
    float sg = holes[h * 3 + 2];
    float inv = 0.5f / (sg * sg);

    float dx = p - x0;
    float dy = p - y0;
    Ex[idx] = __expf(-dx * dx * inv);
    Ey[idx] = __expf(-dy * dy * inv);
}

// ---------------------------------------------------------------------------
// Kernel 2: 64x64 macrotile per wave = 4x4 grid of 16x16 WMMA tiles.
// Per K-step: 4 A-fragment + 4 B-fragment b64 loads feed 16 WMMAs (4x reuse
// of every fragment), cutting L2 operand traffic 4x vs one-tile-per-wave.
//
// f32 WMMA fragment layouts (ISA 7.12.2):
//   A 16x4:  lanes 0-15 -> M=lane, VGPR{0,1}=K{0,1}; lanes 16-31 -> K{2,3}
//   B 4x16:  mirrored with N = lane&15
//   C/D:     VGPR r, lane l -> element [M = r + 8*(l>>4)][N = l&15]
// ---------------------------------------------------------------------------
__global__ __launch_bounds__(256) void gemm_sigmoid_kernel(
    const float* __restrict__ Ex,
    const float* __restrict__ Ey,
    const float* __restrict__ Tptr,
    float* __restrict__ out)
{
    const int lane = threadIdx.x & 31;
    const int wave = threadIdx.x >> 5;
    const int mt   = blockIdx.x * WAVES_PER_BLOCK + wave;
    if (mt >= TOTAL_MACRO) return;     // wave-uniform; EXEC stays all-1s

    const int tm = mt / GRIDT;
    const int tn = mt - tm * GRIDT;

    const int r    = lane & 15;
    const int half = lane >> 4;        // K pair {0,1} vs {2,3}

    // Operand base pointers (edge rows clamped; bogus lanes masked at store).
    const float* pa[4];
    const float* pb[4];
#pragma unroll
    for (int i = 0; i < 4; ++i) {
        int arow = tm * MACRO + i * 16 + r; if (arow > NGRID - 1) arow = NGRID - 1;
        int bcol = tn * MACRO + i * 16 + r; if (bcol > NGRID - 1) bcol = NGRID - 1;
        pa[i] = Ex + arow * NHOLES + 2 * half;
        pb[i] = Ey + bcol * NHOLES + 2 * half;
    }

    v8f acc[4][4] = {};                // 128 VGPRs of accumulators

#pragma unroll
    for (int k = 0; k < KSTEPS; ++k) {
        v2f a[4], b[4];
#pragma unroll
        for (int i = 0; i < 4; ++i) a[i] = *(const v2f*)(pa[i] + k * 4);
#pragma unroll
        for (int i = 0; i < 4; ++i) b[i] = *(const v2f*)(pb[i] + k * 4);
#pragma unroll
        for (int mi = 0; mi < 4; ++mi)
#pragma unroll
            for (int ni = 0; ni < 4; ++ni)
                acc[mi][ni] = __builtin_amdgcn_wmma_f32_16x16x4_f32(
                    false, a[mi], false, b[ni], (short)0, acc[mi][ni],
                    false, false);
    }

    const float invT = __builtin_amdgcn_rcpf(Tptr[0]);
    const int rowBase = tm * MACRO + 8 * half;
    const int colBase = tn * MACRO + r;

    if (tm < GRIDT - 1 && tn < GRIDT - 1) {
        // Interior macrotile: no bounds checks, EXEC stays full.
#pragma unroll
        for (int mi = 0; mi < 4; ++mi)
#pragma unroll
            for (int ni = 0; ni < 4; ++ni) {
                const int col = colBase + ni * 16;
#pragma unroll
                for (int vr = 0; vr < 8; ++vr) {
                    int row = rowBase + mi * 16 + vr;
                    float t = (acc[mi][ni][vr] - 0.5f) * invT;
                    out[row * NGRID + col] =
                        __builtin_amdgcn_rcpf(1.0f + __expf(-t));
                }
            }
    } else {
        // Edge macrotile: predicate stores.
#pragma unroll
        for (int mi = 0; mi < 4; ++mi)
#pragma unroll
            for (int ni = 0; ni < 4; ++ni) {
                const int col = colBase + ni * 16;
#pragma unroll
                for (int vr = 0; vr < 8; ++vr) {
                    int row = rowBase + mi * 16 + vr;
                    if (row < NGRID && col < NGRID) {
                        float t = (acc[mi][ni][vr] - 0.5f) * invT;
                        out[row * NGRID + col] =
                            __builtin_amdgcn_rcpf(1.0f + __expf(-t));
                    }
                }
            }
    }
}

// ---------------------------------------------------------------------------
// Host launch. d_in[0] = hole_position (96x3 f32), d_in[1] = T (1 f32).
// d_out = 4001*4001 f32. d_ws holds Ex/Ey tables (3.07 MB).
// ---------------------------------------------------------------------------
extern "C" void kernel_launch(void* const* d_in, const int* in_sizes, int n_in,
                              void* d_out, int out_size, void* d_ws, size_t ws_size,
                              hipStream_t stream) {
    const float* holes = (const float*)d_in[0];
    const float* Tptr  = (const float*)d_in[1];
    float* out = (float*)d_out;

    float* Ex = (float*)d_ws;
    float* Ey = Ex + (size_t)NGRID * NHOLES;

    int total = NGRID * NHOLES;
    build_tables_kernel<<<(total + 255) / 256, 256, 0, stream>>>(holes, Ex, Ey);

    int blocks = (TOTAL_MACRO + WAVES_PER_BLOCK - 1) / WAVES_PER_BLOCK;
    gemm_sigmoid_kernel<<<blocks, 256, 0, stream>>>(Ex, Ey, Tptr, out);
}